// StripPooling_9483287790272
// MI455X (gfx1250) — compile-verified
//
#include <hip/hip_runtime.h>
#include <hip/hip_bf16.h>
#include <math.h>

// ---------------------------------------------------------------------------
// StripPooling for MI455X (gfx1250, wave32).
// Bandwidth-bound problem (~402 MB HBM traffic @ 23.3 TB/s ~ 17 us).
// Tiny strip-space algebra is folded into two GEMMs run on V_WMMA_F32_16X16X4_F32.
// ---------------------------------------------------------------------------

typedef __attribute__((ext_vector_type(2))) float v2f;
typedef __attribute__((ext_vector_type(8))) float v8f;

#define BN_EPS 1e-5f

// ---------------------------------------------------------------------------
// Kernel 1: strip pooling. One workgroup per (b,c) plane (512 WGs, 256 thr).
// Each plane is 256x256 fp32 (256 KB). hp[b,c,p] = mean of 16-row band,
// vp[b,c,q] = mean of 16-col band. Coalesced float4 loads.
// ---------------------------------------------------------------------------
__global__ __launch_bounds__(256) void pool_kernel(const float* __restrict__ x,
                                                   float* __restrict__ hp,
                                                   float* __restrict__ vp) {
    __shared__ float lds_h[16 * 256];   // per-thread partial per h-bin
    __shared__ float lds_hr[256];
    __shared__ float lds_v[256];

    const int bc = blockIdx.x;              // 0..511 = b*64+c
    const int t  = threadIdx.x;             // 0..255
    const float* plane = x + (size_t)bc * 65536;

    const int r0 = t >> 6;                  // row phase 0..3
    const int c4 = (t & 63) << 2;           // column group (same v-bin for all 4)

    float vsum = 0.f;
    float hacc = 0.f;
    for (int it = 0; it < 64; ++it) {
        const int row = (it << 2) + r0;     // h-bin of row == it>>2 for all r0
        const float4 v = *(const float4*)(plane + row * 256 + c4);
        const float s4 = (v.x + v.y) + (v.z + v.w);
        hacc += s4;
        vsum += s4;
        if ((it & 3) == 3) { lds_h[(it >> 2) * 256 + t] = hacc; hacc = 0.f; }
    }
    lds_v[t] = vsum;
    __syncthreads();

    // reduce hp: 16 bins x 256 partials -> 16 bins x 16 -> 16
    {
        const int bin = t >> 4, j = t & 15;
        float s = 0.f;
        #pragma unroll
        for (int m = 0; m < 16; ++m) s += lds_h[bin * 256 + j * 16 + m];
        lds_hr[bin * 16 + j] = s;
    }
    __syncthreads();
    if (t < 16) {
        float s = 0.f;
        #pragma unroll
        for (int j = 0; j < 16; ++j) s += lds_hr[t * 16 + j];
        hp[bc * 16 + t] = s * (1.0f / 4096.0f);

        float sv = 0.f;
        #pragma unroll
        for (int r = 0; r < 4; ++r)
            #pragma unroll
            for (int j = 0; j < 4; ++j) sv += lds_v[r * 64 + t * 4 + j];
        vp[bc * 16 + t] = sv * (1.0f / 4096.0f);
    }
}

// ---------------------------------------------------------------------------
// Kernel 2: all strip-space algebra. One WG, 256 threads = 8 wave32s.
//   M_h = diag(invF) * (Wf[:, :64] * diag(invH)) * Wh   (64x64, WMMA)
//   M_v = diag(invF) * (Wf[:, 64:] * diag(invV)) * Wv   (64x64, WMMA)
//   gh  = M_h @ hp + d_h    (64 x 128, WMMA)
//   gv  = M_v @ vp + d_v    (64 x 128, WMMA)
// All BN shifts + biases folded into d_h / d_v so the epilogue is just
// sigmoid(gh + gv) * x.
//
// WMMA f32 16x16x4 fragment layout (ISA 7.12.2):
//   A (16x4): lane l holds row M=l%16; VGPR j holds K = 2*(l/16)+j
//   B (4x16): lane l holds col N=l%16; VGPR j holds K = 2*(l/16)+j
//   C/D:      VGPR r: lanes 0-15 -> (M=r, N=lane), lanes 16-31 -> (M=r+8)
// ---------------------------------------------------------------------------
__global__ __launch_bounds__(256) void prep_kernel(
    const float* __restrict__ Wh, const float* __restrict__ bh,
    const float* __restrict__ Wv, const float* __restrict__ bv,
    const float* __restrict__ Wf, const float* __restrict__ bf,
    const float* __restrict__ hg, const float* __restrict__ hb,
    const float* __restrict__ hm, const float* __restrict__ hv,
    const float* __restrict__ vg, const float* __restrict__ vb,
    const float* __restrict__ vm, const float* __restrict__ vvv,
    const float* __restrict__ fg, const float* __restrict__ fb,
    const float* __restrict__ fm, const float* __restrict__ fvv,
    const float* __restrict__ hp, const float* __restrict__ vp,
    float* __restrict__ gh, float* __restrict__ gv) {

    __shared__ float sM[2][64 * 64];    // folded matrices M_h, M_v (32 KB)
    __shared__ float sInvH[64], sInvV[64], sInvF[64];
    __shared__ float sUh[64], sUv[64];
    __shared__ float sDh[64], sDv[64];

    const int t    = threadIdx.x;
    const int lane = t & 31;
    const int wv   = t >> 5;
    const int m16   = lane & 15;
    const int khalf = (lane >> 4) << 1;   // 0 or 2: which K pair this half-wave holds

    // ---- per-channel BN folding -------------------------------------------
    if (t < 64) {
        const float ih  = hg[t] * rsqrtf(hv[t]  + BN_EPS);
        const float iv  = vg[t] * rsqrtf(vvv[t] + BN_EPS);
        const float iff = fg[t] * rsqrtf(fvv[t] + BN_EPS);
        sInvH[t] = ih; sInvV[t] = iv; sInvF[t] = iff;
        sUh[t] = ih * bh[t] + hb[t] - hm[t] * ih;   // affine constant after h-BN
        sUv[t] = iv * bv[t] + vb[t] - vm[t] * iv;   // affine constant after v-BN
    }
    __syncthreads();
    if (t < 64) {
        float ah = 0.f, av = 0.f;
        for (int k = 0; k < 64; ++k) {
            ah += Wf[t * 128 + k]      * sUh[k];
            av += Wf[t * 128 + 64 + k] * sUv[k];
        }
        const float iff = sInvF[t];
        sDh[t] = iff * ah;                                     // gh bias
        sDv[t] = iff * (av + bf[t]) + fb[t] - fm[t] * iff;     // gv bias (+f-BN shift)
    }
    __syncthreads();

    // ---- Stage A: build M_h / M_v with WMMA (32 tiles, 4 per wave) --------
    for (int i = 0; i < 4; ++i) {
        const int tt  = wv * 4 + i;          // wave-uniform tile id
        const int mat = tt >> 4;             // 0 = h, 1 = v
        const int M0  = ((tt >> 2) & 3) << 4;
        const int N0  = (tt & 3) << 4;
        const float* W     = mat ? Wv    : Wh;
        const float* scale = mat ? sInvV : sInvH;
        v8f acc = {};
        for (int s = 0; s < 16; ++s) {       // K = 64 in chunks of 4
            const int kA = (s << 2) + khalf;
            v2f a, b;
            a.x = Wf[(M0 + m16) * 128 + mat * 64 + kA]     * scale[kA];
            a.y = Wf[(M0 + m16) * 128 + mat * 64 + kA + 1] * scale[kA + 1];
            b.x = W[kA       * 64 + N0 + m16];
            b.y = W[(kA + 1) * 64 + N0 + m16];
            acc = __builtin_amdgcn_wmma_f32_16x16x4_f32(
                false, a, false, b, (short)0, acc, false, false);
        }
        #pragma unroll
        for (int r = 0; r < 8; ++r) {
            const int M = M0 + r + ((lane >> 4) << 3);
            sM[mat][M * 64 + N0 + m16] = acc[r] * sInvF[M];
        }
    }
    __syncthreads();

    // ---- Stage B: gh/gv = M @ P + d  (64 x 128 each; 64 tiles, 8/wave) ----
    for (int i = 0; i < 8; ++i) {
        const int tt   = wv * 8 + i;         // wave-uniform tile id
        const int mat  = tt >> 5;            // 0 = h, 1 = v
        const int M0   = ((tt >> 3) & 3) << 4;
        const int bidx = tt & 7;             // batch = N-tile (16 strips per batch)
        const float* P    = mat ? vp  : hp;
        const float* bias = mat ? sDv : sDh;
        float*       G    = mat ? gv  : gh;
        v8f acc = {};
        for (int s = 0; s < 16; ++s) {
            const int kA = (s << 2) + khalf;
            v2f a, b;
            a.x = sM[mat][(M0 + m16) * 64 + kA];
            a.y = sM[mat][(M0 + m16) * 64 + kA + 1];
            b.x = P[bidx * 1024 + kA       * 16 + m16];
            b.y = P[bidx * 1024 + (kA + 1) * 16 + m16];
            acc = __builtin_amdgcn_wmma_f32_16x16x4_f32(
                false, a, false, b, (short)0, acc, false, false);
        }
        #pragma unroll
        for (int r = 0; r < 8; ++r) {
            const int M = M0 + r + ((lane >> 4) << 3);
            G[bidx * 1024 + M * 16 + m16] = acc[r] + bias[M];
        }
    }
}

// ---------------------------------------------------------------------------
// Kernel 3: epilogue. out = sigmoid(gh[b,c,h/16] + gv[b,c,w/16]) * x.
// float4-vectorized; 4 consecutive cols share a w-bin -> one sigmoid per vec4.
// gh/gv are 64 KB total -> stay L2/L0 resident.
// ---------------------------------------------------------------------------
__global__ __launch_bounds__(256) void fuse_kernel(const float4* __restrict__ x,
                                                   const float* __restrict__ gh,
                                                   const float* __restrict__ gv,
                                                   float4* __restrict__ out) {
    const int i  = blockIdx.x * 256 + threadIdx.x;     // exact grid, no tail
    const int w4 = i & 63;
    const int h  = (i >> 6) & 255;
    const int bc = i >> 14;
    const float f = gh[bc * 16 + (h >> 4)] + gv[bc * 16 + (w4 >> 2)];
    const float s = 1.0f / (1.0f + __expf(-f));
    const float4 v = x[i];
    float4 o;
    o.x = v.x * s; o.y = v.y * s; o.z = v.z * s; o.w = v.w * s;
    out[i] = o;
}

// ---------------------------------------------------------------------------
extern "C" void kernel_launch(void* const* d_in, const int* in_sizes, int n_in,
                              void* d_out, int out_size, void* d_ws, size_t ws_size,
                              hipStream_t stream) {
    const float* x  = (const float*)d_in[0];
    const float* Wh = (const float*)d_in[1];
    const float* bh = (const float*)d_in[2];
    const float* Wv = (const float*)d_in[3];
    const float* bv = (const float*)d_in[4];
    const float* Wf = (const float*)d_in[5];
    const float* bf = (const float*)d_in[6];
    const float* hg = (const float*)d_in[7];
    const float* hb = (const float*)d_in[8];
    const float* hm = (const float*)d_in[9];
    const float* hv = (const float*)d_in[10];
    const float* vg = (const float*)d_in[11];
    const float* vb = (const float*)d_in[12];
    const float* vm = (const float*)d_in[13];
    const float* vv = (const float*)d_in[14];
    const float* fg = (const float*)d_in[15];
    const float* fb = (const float*)d_in[16];
    const float* fm = (const float*)d_in[17];
    const float* fv = (const float*)d_in[18];

    float* ws = (float*)d_ws;          // needs 32768 floats = 128 KB
    float* hp = ws;                    // (B, C, 16)
    float* vp = ws + 8192;             // (B, C, 16)
    float* gh = ws + 16384;            // (B, C, 16)
    float* gv = ws + 24576;            // (B, C, 16)

    pool_kernel<<<512, 256, 0, stream>>>(x, hp, vp);
    prep_kernel<<<1, 256, 0, stream>>>(Wh, bh, Wv, bv, Wf, bf,
                                       hg, hb, hm, hv,
                                       vg, vb, vm, vv,
                                       fg, fb, fm, fv,
                                       hp, vp, gh, gv);
    fuse_kernel<<<32768, 256, 0, stream>>>((const float4*)x, gh, gv, (float4*)d_out);
}